// MambaBlock_43696997270090
// MI455X (gfx1250) — compile-verified
//
#include <hip/hip_runtime.h>
#include <hip/hip_bf16.h>

// ---------------------------------------------------------------------------
// Mamba block on MI455X (gfx1250), fp32, wave32.
// GEMMs use V_WMMA_F32_16X16X4_F32 (exact fp32, matches reference; workload is
// bandwidth/latency bound so low-precision WMMA buys nothing here).
// WMMA fragment layouts (ISA 7.12.2, 32-bit):
//   A (16x4, MxK): lane L: M = L&15 ; VGPR v holds K = v + (L>=16 ? 2 : 0)
//   B (4x16, KxN): lane L: N = L&15 ; VGPR v holds K = v + (L>=16 ? 2 : 0)
//   C/D (16x16) :  lane L: N = L&15 ; VGPR r holds M = r + (L>=16 ? 8 : 0)
// => A and B fragments are each a single 8-byte (b64) load when K-chunk start
//    is a multiple of 4 and the row base is 8B aligned.
// The sequential scan (the latency bottleneck) uses CDNA5 async-to-LDS DMA
// (GLOBAL_LOAD_ASYNC_TO_LDS_B128, ASYNCcnt) with double-buffered LDS so the
// next chunk's staging overlaps the current chunk's recurrence.
// ---------------------------------------------------------------------------

#define D_MODEL 128
#define D_INNER 256
#define D_STATE 16
#define D_CONV  4
#define DT_RANK 8
#define BATCH   8
#define SEQ     4096
#define MROWS   (BATCH * SEQ)   // 32768 token rows

typedef __attribute__((ext_vector_type(2))) float v2f;
typedef __attribute__((ext_vector_type(8))) float v8f;

__device__ __forceinline__ v8f wmma4(v2f a, v2f b, v8f c) {
  // (neg_a, A, neg_b, B, c_mod, C, reuse_a, reuse_b)
  return __builtin_amdgcn_wmma_f32_16x16x4_f32(false, a, false, b, (short)0, c,
                                               false, false);
}

__device__ __forceinline__ float silu_f(float z) {
  return z / (1.0f + __expf(-z));
}

// Async DMA: global -> LDS, 16 bytes per lane. LDS dest address is the low
// 32 bits of the generic pointer (flat LDS aperture: addr[31:0] = LDS offset).
__device__ __forceinline__ void async_copy_b128(const float* g, float* l) {
  const unsigned int lds = (unsigned int)(uintptr_t)l;
  const unsigned long long ga = (unsigned long long)(uintptr_t)g;
  asm volatile("global_load_async_to_lds_b128 %0, %1, off"
               :
               : "v"(lds), "v"(ga)
               : "memory");
}

__device__ __forceinline__ void wait_async0() {
#if defined(__has_builtin) && __has_builtin(__builtin_amdgcn_s_wait_asynccnt)
  __builtin_amdgcn_s_wait_asynccnt(0);
#else
  asm volatile("s_wait_asynccnt 0x0" ::: "memory");
#endif
}

// ---------------------------------------------------------------------------
// Kernel 1: in_proj  xz[m, e] = sum_k s[m,k] * W[e,k]
// M = 32768, K = 128, N = 512.  Block = 8 waves; each wave: one 16-row M tile
// x four 16-col N tiles. grid.x = M/16 = 2048.
// ---------------------------------------------------------------------------
__global__ __launch_bounds__(256) void k_in_proj(
    const float* __restrict__ s, const float* __restrict__ W,
    float* __restrict__ xz) {
  const int lane = threadIdx.x & 31;
  const int wave = threadIdx.x >> 5;
  const int m0 = blockIdx.x * 16;
  const int nlo = lane & 15;
  const int koff = (lane >= 16) ? 2 : 0;
  const int n0 = wave * 64;

  const float* arow = s + (size_t)(m0 + nlo) * D_MODEL + koff;
  const float* b0 = W + (size_t)(n0 + 0 + nlo) * D_MODEL + koff;
  const float* b1 = W + (size_t)(n0 + 16 + nlo) * D_MODEL + koff;
  const float* b2 = W + (size_t)(n0 + 32 + nlo) * D_MODEL + koff;
  const float* b3 = W + (size_t)(n0 + 48 + nlo) * D_MODEL + koff;

  v8f acc0 = {}, acc1 = {}, acc2 = {}, acc3 = {};
#pragma unroll 4
  for (int k = 0; k < D_MODEL; k += 4) {
    v2f a = *(const v2f*)(arow + k);
    acc0 = wmma4(a, *(const v2f*)(b0 + k), acc0);
    acc1 = wmma4(a, *(const v2f*)(b1 + k), acc1);
    acc2 = wmma4(a, *(const v2f*)(b2 + k), acc2);
    acc3 = wmma4(a, *(const v2f*)(b3 + k), acc3);
  }

  const int rbase = m0 + ((lane >= 16) ? 8 : 0);
#pragma unroll
  for (int r = 0; r < 8; ++r) {
    size_t o = (size_t)(rbase + r) * (2 * D_INNER);
    xz[o + n0 + 0 + nlo] = acc0[r];
    xz[o + n0 + 16 + nlo] = acc1[r];
    xz[o + n0 + 32 + nlo] = acc2[r];
    xz[o + n0 + 48 + nlo] = acc3[r];
  }
}

// ---------------------------------------------------------------------------
// Kernel 2: causal depthwise conv(4) + bias + SiLU on x = xz[..., :256]
// one thread per (m, d); coalesced along d. grid = M*256/256 = 32768 blocks.
// ---------------------------------------------------------------------------
__global__ __launch_bounds__(256) void k_conv_silu(
    const float* __restrict__ xz, const float* __restrict__ cw,
    const float* __restrict__ cb, float* __restrict__ xact) {
  const int idx = blockIdx.x * 256 + threadIdx.x;
  const int d = idx & (D_INNER - 1);
  const int m = idx >> 8;
  const int l = m & (SEQ - 1);
  float acc = cb[d];
#pragma unroll
  for (int j = 0; j < D_CONV; ++j) {
    const int ls = l - (D_CONV - 1) + j;
    const float v =
        (ls >= 0) ? xz[(size_t)(m - (D_CONV - 1) + j) * (2 * D_INNER) + d]
                  : 0.0f;
    acc += v * cw[d * D_CONV + j];
  }
  xact[(size_t)m * D_INNER + d] = silu_f(acc);
}

// ---------------------------------------------------------------------------
// Kernel 3: x_proj (40 outs/row) + dt_proj (256 outs/row) + softplus.
// Block handles 16 rows staged in LDS. grid = M/16 = 2048.
// ---------------------------------------------------------------------------
__global__ __launch_bounds__(256) void k_xproj_dt(
    const float* __restrict__ xact, const float* __restrict__ xpw,  // (40,256)
    const float* __restrict__ dtw,                                  // (256,8)
    const float* __restrict__ dtb, float* __restrict__ Bc,
    float* __restrict__ Cc, float* __restrict__ dt) {
  __shared__ float sx[16 * D_INNER];
  __shared__ float sdbl[16 * 40];
  const int m0 = blockIdx.x * 16;
  const int tid = threadIdx.x;

  for (int i = tid; i < 16 * D_INNER; i += 256)
    sx[i] = xact[(size_t)m0 * D_INNER + i];
  __syncthreads();

  for (int o = tid; o < 16 * 40; o += 256) {
    const int r = o / 40, j = o - r * 40;
    const float* w = xpw + j * D_INNER;
    const float* xr = sx + r * D_INNER;
    float acc = 0.0f;
    for (int k = 0; k < D_INNER; ++k) acc = fmaf(xr[k], w[k], acc);
    sdbl[r * 40 + j] = acc;
    const int m = m0 + r;
    if (j >= DT_RANK && j < DT_RANK + D_STATE)
      Bc[(size_t)m * D_STATE + (j - DT_RANK)] = acc;
    else if (j >= DT_RANK + D_STATE)
      Cc[(size_t)m * D_STATE + (j - DT_RANK - D_STATE)] = acc;
  }
  __syncthreads();

  // dt[m, tid] = softplus(sum_j x_dbl[m, j] * dtw[tid, j] + dtb[tid])
  const float bias = dtb[tid];
#pragma unroll 4
  for (int r = 0; r < 16; ++r) {
    float acc = bias;
#pragma unroll
    for (int j = 0; j < DT_RANK; ++j)
      acc = fmaf(sdbl[r * 40 + j], dtw[tid * DT_RANK + j], acc);
    const float sp = (acc > 20.0f) ? acc : __logf(1.0f + __expf(acc));
    dt[(size_t)(m0 + r) * D_INNER + tid] = sp;
  }
}

// ---------------------------------------------------------------------------
// Kernel 4: selective scan (the sequential bottleneck) + fused D-skip and
// SiLU(z) gate.  One block per batch, thread = channel d, h[16] in VGPRs.
// Double-buffered LDS staging filled by GLOBAL_LOAD_ASYNC_TO_LDS_B128 so the
// next chunk's DMA overlaps the current chunk's recurrence; one
// s_wait_asynccnt(0) + one barrier per chunk.
// Writes gated y in place of xact (disjoint rows from in-flight prefetch).
// ---------------------------------------------------------------------------
#define SCAN_CH 8
__global__ __launch_bounds__(256) void k_scan(
    const float* __restrict__ dt_g, const float* __restrict__ Bc,
    const float* __restrict__ Cc, const float* __restrict__ xz,
    const float* __restrict__ A_log, const float* __restrict__ D_skip,
    float* __restrict__ xact) {
  __shared__ __align__(16) float sdt[2][SCAN_CH * D_INNER];
  __shared__ __align__(16) float su[2][SCAN_CH * D_INNER];
  __shared__ __align__(16) float sz[2][SCAN_CH * D_INNER];
  __shared__ __align__(16) float sBC[2][SCAN_CH * 32];

  const int b = blockIdx.x;
  const int d = threadIdx.x;

  float Ar[D_STATE];
#pragma unroll
  for (int s2 = 0; s2 < D_STATE; ++s2)
    Ar[s2] = -__expf(A_log[d * D_STATE + s2]);
  const float Dd = D_skip[d];
  float h[D_STATE];
#pragma unroll
  for (int s2 = 0; s2 < D_STATE; ++s2) h[s2] = 0.0f;

  const size_t base = (size_t)b * SEQ;

  // issue async staging for chunk starting at l0 into buffer `buf`
  auto issue_chunk = [&](int buf, int l0) {
    const int NF4 = SCAN_CH * D_INNER / 4;  // 512 float4s per array
    for (int i = d; i < NF4; i += 256) {
      const int el = i * 4;
      const int st = el >> 8;              // el / 256
      const int dd = el & (D_INNER - 1);
      const size_t row = base + l0 + st;
      async_copy_b128(dt_g + row * D_INNER + dd, &sdt[buf][el]);
      async_copy_b128(xact + row * D_INNER + dd, &su[buf][el]);
      async_copy_b128(xz + row * (2 * D_INNER) + D_INNER + dd, &sz[buf][el]);
    }
    const int NBC4 = SCAN_CH * 32 / 4;  // 64 float4s
    for (int i = d; i < NBC4; i += 256) {
      const int el = i * 4;
      const int st = el >> 5;
      const int s4 = el & 31;
      const size_t row = base + l0 + st;
      const float* g = (s4 < D_STATE) ? (Bc + row * D_STATE + s4)
                                      : (Cc + row * D_STATE + (s4 - D_STATE));
      async_copy_b128(g, &sBC[buf][el]);
    }
  };

  issue_chunk(0, 0);

  const int NCHUNK = SEQ / SCAN_CH;
  for (int c = 0; c < NCHUNK; ++c) {
    // current chunk's async loads (issued last iteration) must be complete in
    // LDS for all waves before anyone reads it.
    wait_async0();
    __syncthreads();

    // overlap: prefetch chunk c+1 into the other buffer while computing c.
    // buf[c&1] is only rewritten by issue(c+2), which runs after iteration
    // c+1's barrier, i.e. after every wave finished computing chunk c.
    if (c + 1 < NCHUNK) issue_chunk((c + 1) & 1, (c + 1) * SCAN_CH);

    const int p = c & 1;
    const int l0 = c * SCAN_CH;
    for (int st = 0; st < SCAN_CH; ++st) {
      const float dtv = sdt[p][st * D_INNER + d];
      const float u = su[p][st * D_INNER + d];
      const float du = dtv * u;
      const float* bc = &sBC[p][st * 32];
      float y = 0.0f;
#pragma unroll
      for (int s2 = 0; s2 < D_STATE; ++s2) {
        const float dA = __expf(dtv * Ar[s2]);
        h[s2] = fmaf(dA, h[s2], du * bc[s2]);
        y = fmaf(h[s2], bc[D_STATE + s2], y);
      }
      const float z = sz[p][st * D_INNER + d];
      xact[(base + l0 + st) * D_INNER + d] = (y + u * Dd) * silu_f(z);
    }
  }
}

// ---------------------------------------------------------------------------
// Kernel 5: out_proj  h[m, o] = sum_k y'[m,k] * W[o,k]
// M = 32768, K = 256, N = 128.  8 waves, wave = one 16x16 N tile. grid = 2048.
// ---------------------------------------------------------------------------
__global__ __launch_bounds__(256) void k_out_proj(
    const float* __restrict__ yp, const float* __restrict__ W,  // (128,256)
    float* __restrict__ hbuf) {
  const int lane = threadIdx.x & 31;
  const int wave = threadIdx.x >> 5;
  const int m0 = blockIdx.x * 16;
  const int nlo = lane & 15;
  const int koff = (lane >= 16) ? 2 : 0;
  const int n0 = wave * 16;

  const float* arow = yp + (size_t)(m0 + nlo) * D_INNER + koff;
  const float* brow = W + (size_t)(n0 + nlo) * D_INNER + koff;

  v8f acc = {};
#pragma unroll 8
  for (int k = 0; k < D_INNER; k += 4)
    acc = wmma4(*(const v2f*)(arow + k), *(const v2f*)(brow + k), acc);

  const int rbase = m0 + ((lane >= 16) ? 8 : 0);
#pragma unroll
  for (int r = 0; r < 8; ++r)
    hbuf[(size_t)(rbase + r) * D_MODEL + n0 + nlo] = acc[r];
}

// ---------------------------------------------------------------------------
// Kernel 6: LayerNorm (in LDS, 132-float padded rows -> 8B-aligned ds_load_b64
// A fragments) + fc GEMM (K=128, N=128) + ReLU.  grid = 2048.
// ---------------------------------------------------------------------------
#define LNP 132  // row pitch in floats (528B, keeps b64 alignment)
__global__ __launch_bounds__(256) void k_ln_fc(
    const float* __restrict__ hbuf, const float* __restrict__ gamma,
    const float* __restrict__ beta, const float* __restrict__ fcw,  // (128,128)
    float* __restrict__ out) {
  __shared__ float sh[16 * LNP];
  __shared__ float sred[16 * 16];
  __shared__ float sred2[16 * 16];
  __shared__ float smu[16];
  __shared__ float srs[16];

  const int m0 = blockIdx.x * 16;
  const int tid = threadIdx.x;
  const int r = tid >> 4;
  const int seg = tid & 15;

  // load + partial sums
  {
    float s1 = 0.0f, s2 = 0.0f;
#pragma unroll
    for (int j = 0; j < 8; ++j) {
      const float v = hbuf[(size_t)(m0 + r) * D_MODEL + seg * 8 + j];
      sh[r * LNP + seg * 8 + j] = v;
      s1 += v;
      s2 += v * v;
    }
    sred[r * 16 + seg] = s1;
    sred2[r * 16 + seg] = s2;
  }
  __syncthreads();
  if (tid < 16) {
    float s1 = 0.0f, s2 = 0.0f;
    for (int i = 0; i < 16; ++i) {
      s1 += sred[tid * 16 + i];
      s2 += sred2[tid * 16 + i];
    }
    const float mu = s1 * (1.0f / D_MODEL);
    const float var = s2 * (1.0f / D_MODEL) - mu * mu;
    smu[tid] = mu;
    srs[tid] = rsqrtf(var + 1e-5f);
  }
  __syncthreads();
  {
    const float mu = smu[r], rs = srs[r];
#pragma unroll
    for (int j = 0; j < 8; ++j) {
      const int c = seg * 8 + j;
      sh[r * LNP + c] = (sh[r * LNP + c] - mu) * rs * gamma[c] + beta[c];
    }
  }
  __syncthreads();

  // WMMA over normalized rows: wave w -> N tile w (cols w*16..w*16+15)
  const int lane = tid & 31;
  const int wave = tid >> 5;
  const int nlo = lane & 15;
  const int koff = (lane >= 16) ? 2 : 0;
  const int n0 = wave * 16;

  const float* als = sh + nlo * LNP + koff;                       // LDS
  const float* brow = fcw + (size_t)(n0 + nlo) * D_MODEL + koff;  // global

  v8f acc = {};
#pragma unroll 8
  for (int k = 0; k < D_MODEL; k += 4)
    acc = wmma4(*(const v2f*)(als + k), *(const v2f*)(brow + k), acc);

  const int rbase = m0 + ((lane >= 16) ? 8 : 0);
#pragma unroll
  for (int rr = 0; rr < 8; ++rr)
    out[(size_t)(rbase + rr) * D_MODEL + n0 + nlo] = fmaxf(acc[rr], 0.0f);
}

// ---------------------------------------------------------------------------
extern "C" void kernel_launch(void* const* d_in, const int* in_sizes, int n_in,
                              void* d_out, int out_size, void* d_ws,
                              size_t ws_size, hipStream_t stream) {
  const float* s = (const float*)d_in[0];
  const float* in_proj_w = (const float*)d_in[1];
  const float* conv_w = (const float*)d_in[2];
  const float* conv_b = (const float*)d_in[3];
  const float* x_proj_w = (const float*)d_in[4];
  const float* dt_proj_w = (const float*)d_in[5];
  const float* dt_proj_b = (const float*)d_in[6];
  const float* A_log = (const float*)d_in[7];
  const float* D_skip = (const float*)d_in[8];
  const float* out_proj_w = (const float*)d_in[9];
  const float* ln_gamma = (const float*)d_in[10];
  const float* ln_beta = (const float*)d_in[11];
  const float* fc_w = (const float*)d_in[12];

  float* ws = (float*)d_ws;
  float* xz = ws;                                  // M * 512
  float* xact = xz + (size_t)MROWS * 2 * D_INNER;  // M * 256
  float* dt = xact + (size_t)MROWS * D_INNER;      // M * 256
  float* Bc = dt + (size_t)MROWS * D_INNER;        // M * 16
  float* Cc = Bc + (size_t)MROWS * D_STATE;        // M * 16
  float* hbuf = dt;  // dt dead after k_scan; reuse for out_proj result (M*128)

  k_in_proj<<<MROWS / 16, 256, 0, stream>>>(s, in_proj_w, xz);
  k_conv_silu<<<(MROWS * D_INNER) / 256, 256, 0, stream>>>(xz, conv_w, conv_b,
                                                           xact);
  k_xproj_dt<<<MROWS / 16, 256, 0, stream>>>(xact, x_proj_w, dt_proj_w,
                                             dt_proj_b, Bc, Cc, dt);
  k_scan<<<BATCH, 256, 0, stream>>>(dt, Bc, Cc, xz, A_log, D_skip, xact);
  k_out_proj<<<MROWS / 16, 256, 0, stream>>>(xact, out_proj_w, hbuf);
  k_ln_fc<<<MROWS / 16, 256, 0, stream>>>(hbuf, ln_gamma, ln_beta, fc_w,
                                          (float*)d_out);
}